// CAM_Module_Cross_26585847562851
// MI455X (gfx1250) — compile-verified
//
#include <hip/hip_runtime.h>
#include <hip/hip_bf16.h>

typedef __attribute__((ext_vector_type(2))) float v2f;
typedef __attribute__((ext_vector_type(8))) float v8f;

#define Cc 64
#define Hh 64
#define Ww 64
#define Ff 10
#define Gg 5
#define KP 12      // K=10 padded to 12 for 3 x (16x16x4) WMMA k-steps
#define EPITCH 65  // energy row pitch (bank-conflict pad)

__global__ __launch_bounds__(512) void cam_pixel_kernel(
    const float* __restrict__ x,
    const float* __restrict__ pv,
    const float* __restrict__ w1, const float* __restrict__ b1,
    const float* __restrict__ w2, const float* __restrict__ b2,
    float* __restrict__ out)
{
    __shared__ float xp[Cc][25];       // 5x5 zero-padded patch per C-slice
    __shared__ float y1[Cc][Gg * 9];   // relu(conv1) at 3x3 positions
    __shared__ float Ml[Cc][KP];       // per-pixel feature matrix M[64][10] (+pad)
    __shared__ float en[Cc][EPITCH];   // energy = M M^T
    __shared__ float wbuf[510];        // w1(45) b1(5) w2(450) b2(10)
    __shared__ float vbuf[Cc];         // proj_value column

    const int tid = threadIdx.x;
    const int p  = blockIdx.x;          // pixel id in [0, 16384)
    const int b  = p >> 12;
    const int hw = p & 4095;
    const int h  = hw >> 6;
    const int w  = hw & 63;

    // ---- stage weights + value vector into LDS ----
    if (tid < 45)       wbuf[tid] = w1[tid];
    else if (tid < 50)  wbuf[tid] = b1[tid - 45];
    else if (tid < 500) wbuf[tid] = w2[tid - 50];
    else if (tid < 510) wbuf[tid] = b2[tid - 500];

    if (tid < Cc) vbuf[tid] = pv[((b * Cc + tid) * Hh + h) * Ww + w];

    // ---- 5x5 x-patch per channel, zero padded at image border ----
    for (int idx = tid; idx < Cc * 25; idx += 512) {
        int c = idx / 25, i = idx % 25;
        int hh = h + (i / 5) - 2;
        int ww = w + (i % 5) - 2;
        float v = 0.f;
        if ((unsigned)hh < (unsigned)Hh && (unsigned)ww < (unsigned)Ww)
            v = x[((b * Cc + c) * Hh + hh) * Ww + ww];
        xp[c][i] = v;
    }
    __syncthreads();

    // ---- conv1 (1->5, 3x3) + ReLU, evaluated at the 3x3 neighborhood ----
    for (int idx = tid; idx < Cc * Gg * 9; idx += 512) {
        int c = idx / 45, rem = idx % 45;
        int g = rem / 9, pos = rem % 9;
        int ph = pos / 3, pw = pos % 3;
        float acc = wbuf[45 + g];
        const float* wg = &wbuf[g * 9];
        #pragma unroll
        for (int kh = 0; kh < 3; ++kh)
            #pragma unroll
            for (int kw = 0; kw < 3; ++kw)
                acc += wg[kh * 3 + kw] * xp[c][(ph + kh) * 5 + (pw + kw)];
        y1[c][g * 9 + pos] = fmaxf(acc, 0.f);
    }
    __syncthreads();

    // ---- conv2 (5->10, 3x3) -> t[f][c], relayout to M[c'][f'] (torch reshape quirk) ----
    for (int idx = tid; idx < Ff * Cc; idx += 512) {
        int f = idx >> 6, c = idx & 63;
        float acc = wbuf[500 + f];
        const float* wf = &wbuf[50 + f * 45];
        #pragma unroll
        for (int gp = 0; gp < 45; ++gp)
            acc += wf[gp] * y1[c][gp];
        int flat = f * Cc + c;            // flat = f*64 + c = c'*10 + f'
        Ml[flat / Ff][flat % Ff] = acc;
    }
    if (tid < Cc * 2) Ml[tid >> 1][Ff + (tid & 1)] = 0.f;   // K padding
    __syncthreads();

    // ---- energy = M M^T via V_WMMA_F32_16X16X4_F32: 16 waves, one 16x16 tile each ----
    {
        const int wave = tid >> 5;
        const int lane = tid & 31;
        const int ti = wave >> 2, tj = wave & 3;     // output tile (row, col)
        const int r  = lane & 15;
        const int kh = (lane < 16) ? 0 : 2;          // A/B: lanes 0-15 -> K=k0,k0+1; 16-31 -> k0+2,k0+3
        v8f acc = {0.f, 0.f, 0.f, 0.f, 0.f, 0.f, 0.f, 0.f};
        #pragma unroll
        for (int k0 = 0; k0 < KP; k0 += 4) {
            v2f a, bb;
            a.x  = Ml[ti * 16 + r][k0 + kh];
            a.y  = Ml[ti * 16 + r][k0 + kh + 1];
            bb.x = Ml[tj * 16 + r][k0 + kh];         // B = M^T: row of the column tile
            bb.y = Ml[tj * 16 + r][k0 + kh + 1];
            acc = __builtin_amdgcn_wmma_f32_16x16x4_f32(
                false, a, false, bb, (short)0, acc, false, false);
        }
        const int mbase = ti * 16 + ((lane < 16) ? 0 : 8);   // D: VGPR i -> rows i / i+8
        const int col   = tj * 16 + r;
        #pragma unroll
        for (int i = 0; i < 8; ++i)
            en[mbase + i][col] = acc[i];
    }
    __syncthreads();

    // ---- row softmax fused with att @ v (8 threads per row) ----
    {
        const int row = tid >> 3;
        const int l8  = tid & 7;
        const int j0  = l8 * 8;
        float e[8];
        float mx = -INFINITY;
        #pragma unroll
        for (int i = 0; i < 8; ++i) { e[i] = en[row][j0 + i]; mx = fmaxf(mx, e[i]); }
        #pragma unroll
        for (int off = 1; off < 8; off <<= 1)
            mx = fmaxf(mx, __shfl_xor(mx, off, 8));
        float s = 0.f, dot = 0.f;
        #pragma unroll
        for (int i = 0; i < 8; ++i) {
            float ex = __expf(e[i] - mx);
            s   += ex;
            dot += ex * vbuf[j0 + i];
        }
        #pragma unroll
        for (int off = 1; off < 8; off <<= 1) {
            s   += __shfl_xor(s, off, 8);
            dot += __shfl_xor(dot, off, 8);
        }
        if (l8 == 0)
            out[((b * Cc + row) * Hh + h) * Ww + w] = dot / s;
    }
}

extern "C" void kernel_launch(void* const* d_in, const int* in_sizes, int n_in,
                              void* d_out, int out_size, void* d_ws, size_t ws_size,
                              hipStream_t stream) {
    (void)in_sizes; (void)n_in; (void)out_size; (void)d_ws; (void)ws_size;
    const float* x  = (const float*)d_in[0];
    const float* pv = (const float*)d_in[1];
    const float* w1 = (const float*)d_in[2];
    const float* b1 = (const float*)d_in[3];
    const float* w2 = (const float*)d_in[4];
    const float* b2 = (const float*)d_in[5];
    float* out = (float*)d_out;

    // one block per pixel: 4 * 64 * 64 = 16384 blocks, 16 waves each
    cam_pixel_kernel<<<4 * 64 * 64, 512, 0, stream>>>(x, pv, w1, b1, w2, b2, out);
}